// SparsemaxBisect_55327768707606
// MI455X (gfx1250) — compile-verified
//
#include <hip/hip_runtime.h>
#include <float.h>
#include <stdint.h>

// Problem shape (from reference): X[4096][32000] fp32, row-wise sparsemax (alpha=2) by bisection.
#define COLS      32000
#define COLS4     8000          // float4 per row
#define TPB       1024          // 32 wave32 waves -> fills one WGP (8 waves per SIMD32)
#define NWAVES    (TPB / 32)
#define VPT       8             // float4 per thread -> 32 fp32 in VGPRs per thread
#define PADF      (TPB * VPT * 4)  // 32768 floats of LDS row storage (row + -inf pad)
#define N_ITER    50

typedef __attribute__((ext_vector_type(4))) unsigned int u32x4;
typedef __attribute__((ext_vector_type(8))) int          i32x8;
typedef __attribute__((ext_vector_type(4))) int          i32x4;

// One ds_add_f32 per wave into a fresh pre-zeroed slot, one barrier, broadcast read.
__device__ __forceinline__ float block_sum(float v, float* slot, int lane) {
#pragma unroll
  for (int off = 16; off >= 1; off >>= 1) v += __shfl_xor(v, off);
  if (lane == 0) atomicAdd(slot, v);          // lowers to ds_add_f32 (no-return)
  __syncthreads();
  return *slot;                               // LDS broadcast load
}

__global__ __launch_bounds__(TPB)
void SparsemaxBisect_kernel(const float* __restrict__ X, float* __restrict__ Y, int rows) {
  __shared__ float row[PADF];                 // 128 KB row staging (fits in 320 KB WGP LDS)
  __shared__ float red_max[NWAVES];
  __shared__ float red_sum[N_ITER + 2];       // slot 0: f_lo pass; slots 1..50: bisect passes

  const int tid  = threadIdx.x;
  const int wave = tid >> 5;
  const int lane = tid & 31;
  const int r    = blockIdx.x;
  if (r >= rows) return;

  const float* __restrict__ Xrow = X + (size_t)r * COLS;
  float4* __restrict__ Y4 = (float4*)(Y + (size_t)r * COLS);

  if (tid < N_ITER + 2) red_sum[tid] = 0.0f;  // pre-zero all reduction slots once

  float4 v[VPT];

#if __has_builtin(__builtin_amdgcn_tensor_load_to_lds)
  // ---- CDNA5 TDM: DMA the whole 128 KB row global->LDS with one descriptor ----
  if (wave == 0) {
    const uint32_t lds_base = (uint32_t)(uintptr_t)(void*)row;   // low 32b of shared addr = LDS byte offset
    const uint64_t gaddr    = (uint64_t)(uintptr_t)Xrow;
    // D# group 0: count=1 | lds_addr | global_addr[56:0] | type=2 ("image")
    u32x4 g0 = { 1u,
                 lds_base,
                 (uint32_t)(gaddr & 0xFFFFFFFFull),
                 (uint32_t)((gaddr >> 32) & 0x01FFFFFFull) | (2u << 30) };
    // D# group 1: data_size=4B; tensor_dim0=32000, tensor_dim1=1; tile=32000x1; dim0_stride=32000
    i32x8 g1 = { (int)(2u << 16),                        // [17:16] data_size=2 (4 bytes)
                 (int)(((unsigned)COLS & 0xFFFFu) << 16),// tensor_dim0[15:0] -> bits[63:48]
                 (int)(((unsigned)COLS >> 16) | (1u << 16)), // tensor_dim0[31:16] | tensor_dim1=1
                 (int)((unsigned)COLS << 16),            // tile_dim0=32000 -> bits[127:112]
                 1,                                      // tile_dim1=1
                 COLS,                                   // tensor_dim0_stride[31:0]
                 (int)(((unsigned)COLS & 0xFFFFu) << 16),// tensor_dim1_stride[15:0] -> bits[223:208]
                 0 };
    i32x4 gz4 = { 0, 0, 0, 0 };                          // groups 2/3 unused (<=2D tensor)
    i32x8 gz8 = { 0, 0, 0, 0, 0, 0, 0, 0 };              // extra group (6-arg toolchain form)
    __builtin_amdgcn_tensor_load_to_lds(g0, g1, gz4, gz4, gz8, /*cpol*/0);
    __builtin_amdgcn_s_wait_tensorcnt(0);                // TENSORcnt==0 before releasing barrier
  }
  // Pad tail [32000..32768) with -FLT_MAX so register gather & max are branch-free.
  if (tid < (PADF - COLS)) row[COLS + tid] = -FLT_MAX;
  __syncthreads();                                       // DMA + pad + slot-init visible to all waves
  // Conflict-free LDS->VGPR gather (ds_load_b128, consecutive lanes -> consecutive banks)
#pragma unroll
  for (int j = 0; j < VPT; ++j)
    v[j] = ((const float4*)row)[j * TPB + tid];
#else
  // Fallback: direct coalesced global->VGPR loads.
  const float4* __restrict__ X4 = (const float4*)Xrow;
#pragma unroll
  for (int j = 0; j < VPT; ++j) {
    const int idx = j * TPB + tid;
    if (idx < COLS4) v[j] = X4[idx];
    else             v[j] = make_float4(-FLT_MAX, -FLT_MAX, -FLT_MAX, -FLT_MAX);
  }
  __syncthreads();                                       // slot-init visible
#endif

  // ---- Pass 0: row max ----
  float m = -FLT_MAX;
#pragma unroll
  for (int j = 0; j < VPT; ++j)
    m = fmaxf(m, fmaxf(fmaxf(v[j].x, v[j].y), fmaxf(v[j].z, v[j].w)));
#pragma unroll
  for (int off = 16; off >= 1; off >>= 1) m = fmaxf(m, __shfl_xor(m, off));
  if (lane == 0) red_max[wave] = m;
  __syncthreads();
  float mv = red_max[0];
#pragma unroll
  for (int i = 1; i < NWAVES; ++i) mv = fmaxf(mv, red_max[i]);

  // ---- Pass 1: f_lo at tau_lo = max - 1 ----
  float tau_lo = mv - 1.0f;
  const float tau_hi = mv - (1.0f / (float)COLS);
  float dm = tau_hi - tau_lo;
  float part = 0.0f;
#pragma unroll
  for (int j = 0; j < VPT; ++j) {
    part += fmaxf(v[j].x - tau_lo, 0.0f);
    part += fmaxf(v[j].y - tau_lo, 0.0f);
    part += fmaxf(v[j].z - tau_lo, 0.0f);
    part += fmaxf(v[j].w - tau_lo, 0.0f);
  }
  const float f_lo = block_sum(part, &red_sum[0], lane) - 1.0f;

  // ---- 50 bisection iterations, all data register-resident ----
  float tau_m = tau_lo;
  float f_m   = f_lo;
  for (int it = 0; it < N_ITER; ++it) {
    dm *= 0.5f;
    tau_m = tau_lo + dm;
    float p = 0.0f;
#pragma unroll
    for (int j = 0; j < VPT; ++j) {
      p += fmaxf(v[j].x - tau_m, 0.0f);
      p += fmaxf(v[j].y - tau_m, 0.0f);
      p += fmaxf(v[j].z - tau_m, 0.0f);
      p += fmaxf(v[j].w - tau_m, 0.0f);
    }
    f_m = block_sum(p, &red_sum[1 + it], lane) - 1.0f;   // one barrier per iteration
    if (f_m * f_lo >= 0.0f) tau_lo = tau_m;              // uniform update
  }

  // ---- Final: p = clip(x - tau_m, 0) / sum; sum == f_m + 1 from last pass ----
  const float inv = 1.0f / (f_m + 1.0f);
#pragma unroll
  for (int j = 0; j < VPT; ++j) {
    const int idx = j * TPB + tid;
    if (idx < COLS4) {
      float4 o;
      o.x = fmaxf(v[j].x - tau_m, 0.0f) * inv;
      o.y = fmaxf(v[j].y - tau_m, 0.0f) * inv;
      o.z = fmaxf(v[j].z - tau_m, 0.0f) * inv;
      o.w = fmaxf(v[j].w - tau_m, 0.0f) * inv;
      Y4[idx] = o;                                       // coalesced global_store_b128
    }
  }
}

extern "C" void kernel_launch(void* const* d_in, const int* in_sizes, int n_in,
                              void* d_out, int out_size, void* d_ws, size_t ws_size,
                              hipStream_t stream) {
  const float* X = (const float*)d_in[0];
  float*       Y = (float*)d_out;
  const int rows = in_sizes[0] / COLS;                   // 4096
  SparsemaxBisect_kernel<<<rows, TPB, 0, stream>>>(X, Y, rows);
}